// PyrmaidCostVolume_36301063586246
// MI455X (gfx1250) — compile-verified
//
#include <hip/hip_runtime.h>
#include <stdint.h>

// Problem dims (fixed by the harness/reference).
#define BD   16
#define DD   64
#define HH   128
#define WW   240
#define NLEV 3
#define NS   10   // SAMPLE_POINTS + 1

typedef unsigned int v4u __attribute__((ext_vector_type(4)));
typedef int          v8i __attribute__((ext_vector_type(8)));
typedef int          v4i __attribute__((ext_vector_type(4)));

// Fetch pyramid level value at (d, w) by averaging level-0 rows held in LDS.
__device__ __forceinline__ float fetch_lvl(const float* lds0, int lev, int d, int w) {
    if (lev == 0) {
        return lds0[d * WW + w];
    } else if (lev == 1) {
        return 0.5f * (lds0[(2 * d) * WW + w] + lds0[(2 * d + 1) * WW + w]);
    } else {
        return 0.25f * (lds0[(4 * d) * WW + w] + lds0[(4 * d + 1) * WW + w] +
                        lds0[(4 * d + 2) * WW + w] + lds0[(4 * d + 3) * WW + w]);
    }
}

__global__ __launch_bounds__(256)
void pyramid_cost_volume_kernel(const float* __restrict__ cv,
                                const int*   __restrict__ radius_p,
                                const float* __restrict__ cur_disp,
                                float*       __restrict__ out) {
    // 64 rows x 240 cols of f32 = 60 KB LDS: the full D-column tile for (b, h).
    __shared__ float lds0[DD * WW];

    const int tid = threadIdx.x;
    const int bh  = blockIdx.x;
    const int b   = bh / HH;
    const int h   = bh - b * HH;

    // --- TDM: one 2D descriptor moves cv[b, 0:64, h, 0:240] -> LDS ---------
    // Tile: dim0 = 240 contiguous f32 (W), dim1 = 64 rows (D) with stride
    // H*W = 30720 elements. DMA writes straight to LDS; tracked by TENSORcnt.
    if (tid == 0) {
        uint64_t ga = (uint64_t)(uintptr_t)cv +
                      ((uint64_t)b * (uint64_t)(DD * HH * WW) +
                       (uint64_t)h * (uint64_t)WW) * 4ull;
        // Generic LDS pointer: low 32 bits are the LDS byte offset.
        uint32_t la = (uint32_t)(uintptr_t)(&lds0[0]);

        v4u g0;
        g0.x = 1u;                                   // count=1 (valid), user mode
        g0.y = la;                                   // lds_addr (bytes)
        g0.z = (uint32_t)ga;                         // global_addr[31:0]
        g0.w = (uint32_t)((ga >> 32) & 0x1FFFFFFu)   // global_addr[56:32]
               | (2u << 30);                         // type = 2 ("image")

        v8i g1;
        g1[0] = (int)(2u << 16);                     // data_size = 2 -> 4 bytes
        g1[1] = (int)((unsigned)WW << 16);           // tensor_dim0 = 240 (bits 63:48)
        g1[2] = (int)((unsigned)DD << 16);           // tensor_dim1 = 64  (bits 95:80)
        g1[3] = (int)((unsigned)WW << 16);           // tile_dim0   = 240 (bits 127:112)
        g1[4] = DD;                                  // tile_dim1 = 64, tile_dim2 = 0
        g1[5] = HH * WW;                             // tensor_dim0_stride = 30720
        g1[6] = 0;                                   // stride hi / dim1_stride lo
        g1[7] = 0;                                   // dim1_stride hi

        v4i g2 = {0, 0, 0, 0};                       // 2D tensor: groups 2/3 unused
        v4i g3 = {0, 0, 0, 0};

#if __has_include(<hip/amd_detail/amd_gfx1250_TDM.h>)
        v8i gx = {0, 0, 0, 0, 0, 0, 0, 0};           // amdgpu-toolchain: 6-arg form
        __builtin_amdgcn_tensor_load_to_lds(g0, g1, g2, g3, gx, 0);
#else
        __builtin_amdgcn_tensor_load_to_lds(g0, g1, g2, g3, 0);   // ROCm 7.2: 5-arg
#endif
        __builtin_amdgcn_s_wait_tensorcnt(0);
    }
    __syncthreads();

    // --- Per-pixel sampling (thread tid owns pixel w = tid) ----------------
    if (tid < WW) {
        const int   w        = tid;
        const float radius   = (float)(*radius_p);
        const float interval = (2.0f * radius) / 9.0f;     // (upper-lower)/S
        const float disp     = cur_disp[((size_t)b * HH + h) * WW + w];

        #pragma unroll
        for (int lev = 0; lev < NLEV; ++lev) {
            const int   Dl    = DD >> lev;
            const float refd  = disp * (lev == 0 ? 1.0f : (lev == 1 ? 0.5f : 0.25f));
            const float lower = refd - radius;
            const float dmax  = (float)(Dl - 1);

            #pragma unroll
            for (int s = 0; s < NS; ++s) {
                const float cand = lower + (float)s * interval;
                float cc = __builtin_ceilf(cand);
                float fc = __builtin_floorf(cand);
                cc = fminf(fmaxf(cc, 0.0f), dmax);
                fc = fminf(fmaxf(fc, 0.0f), dmax);
                const float floor_rate = cc - cand;       // clip-then-rate (STE fwd)
                const float ceil_rate  = 1.0f - floor_rate;
                const int   ic = (int)cc;
                const int   ifl = (int)fc;

                const float vc = fetch_lvl(lds0, lev, ic,  w);
                const float vf = fetch_lvl(lds0, lev, ifl, w);

                const size_t oidx =
                    (((size_t)b * (NLEV * NS) + (size_t)(lev * NS + s)) * HH + h) * WW + w;
                out[oidx] = vc * ceil_rate + vf * floor_rate;
            }
        }
    }
}

extern "C" void kernel_launch(void* const* d_in, const int* in_sizes, int n_in,
                              void* d_out, int out_size, void* d_ws, size_t ws_size,
                              hipStream_t stream) {
    const float* cv       = (const float*)d_in[0];  // cost_volume [16,64,128,240] f32
    const int*   radius   = (const int*)  d_in[1];  // scalar int
    const float* cur_disp = (const float*)d_in[2];  // [16,1,128,240] f32
    float*       out      = (float*)d_out;          // [16,30,128,240] f32

    dim3 grid(BD * HH);   // one block per (b, h): 2048 blocks
    dim3 block(256);      // 8 wave32s
    hipLaunchKernelGGL(pyramid_cost_volume_kernel, grid, block, 0, stream,
                       cv, radius, cur_disp, out);
}